// BandStopFilter_14559939133813
// MI455X (gfx1250) — compile-verified
//
#include <hip/hip_runtime.h>
#include <hip/hip_bf16.h>
#include <stdint.h>

// Problem geometry (matches reference)
#define NLEN   131072            // samples per row
#define NROWS  32                // 4*8 rows
#define PAD    27                // filtfilt reflective pad
#define NEXT   (NLEN + 2*PAD)    // 131126 extended samples
#define ORD    8                 // IIR state dimension (order-8 filter)
#define NC     9                 // number of b/a taps

// Chunk-parallel scan geometry
#define NTH    256               // threads per block = chunks per row
#define CHUNK  516               // samples per chunk (L); NTH*CHUNK >= NEXT
#define TOT    (NTH*CHUNK)       // 132096 padded samples per row
#define TILE   4                 // steps staged per async tile (CHUNK % TILE == 0)
#define NTILES (CHUNK/TILE)      // 129

// ---------------------------------------------------------------------------
// CDNA5 async global->LDS staging (ASYNCcnt-tracked, gfx1250-only path)
// ---------------------------------------------------------------------------
__device__ __forceinline__ void async_ld_b64(uint32_t lds_off, const double* gptr) {
  // vdst = VGPR holding LDS byte address, vaddr = 64-bit global address, no SADDR
  asm volatile("global_load_async_to_lds_b64 %0, %1, off"
               :: "v"(lds_off), "v"(gptr) : "memory");
}

// ---------------------------------------------------------------------------
// Kernel 1: build reflect-extended signal in f64, transposed layout
//   extT[row][s*NTH + t] = ext_row[t*CHUNK + s]  (0 beyond NEXT)
// so that the scan kernels read/write with lane-consecutive addresses.
// ---------------------------------------------------------------------------
__global__ __launch_bounds__(256) void build_extT(const float* __restrict__ x,
                                                  double* __restrict__ extT) {
  size_t idx = (size_t)blockIdx.x * 256 + threadIdx.x;
  if (idx >= (size_t)NROWS * TOT) return;
  int r = (int)(idx / TOT);
  int p = (int)(idx % TOT);
  int s = p / NTH;
  int t = p % NTH;
  long k = (long)t * CHUNK + s;          // sample index in extended signal
  double v = 0.0;
  if (k < NEXT) {
    const float* xr = x + (size_t)r * NLEN;
    if (k < PAD) {
      v = 2.0 * (double)xr[0] - (double)xr[PAD - k];
    } else if (k < PAD + NLEN) {
      v = (double)xr[k - PAD];
    } else {
      long j = k - (PAD + NLEN);
      v = 2.0 * (double)xr[NLEN - 1] - (double)xr[NLEN - 2 - j];
    }
  }
  extT[idx] = v;
}

// ---------------------------------------------------------------------------
// Chunk scan body: runs CHUNK recurrence steps for this thread's chunk,
// streaming input through a double-buffered async LDS pipeline.
//   mode 0: input = extT (already zero-padded, direct layout)
//   mode 1: input = y1T read time-reversed; samples past NEXT are zero
// WRITE=false: state-only pass.  WRITE=true: emit outputs.
// ---------------------------------------------------------------------------
template<bool WRITE>
__device__ __forceinline__ void run_chunk(const double* __restrict__ inRow,
                                          double* __restrict__ outRowT,
                                          float* __restrict__ outF,
                                          int r, int t, int mode,
                                          const double* an, const double* bn,
                                          double* z,
                                          double (*tileS)[TILE * NTH]) {
  const uint32_t base = (uint32_t)(uint64_t)(uintptr_t)(&tileS[0][0]);

  auto issue_tile = [&](int tl, int buf) {
    #pragma unroll
    for (int j = 0; j < TILE; ++j) {
      int s = tl * TILE + j;
      const double* g;
      if (mode == 0) {
        g = inRow + (size_t)s * NTH + t;                  // coalesced across lanes
      } else {
        int q  = NEXT - 1 - s;
        int kk = q - t * CHUNK;                           // reversed sample index
        if (kk >= 0)
          g = inRow + (size_t)(q % CHUNK) * NTH + (q / CHUNK - t); // lane-consecutive
        else
          g = inRow;                                      // clamped; masked at consume
      }
      uint32_t off = base + (uint32_t)(((buf * TILE + j) * NTH + t) * 8);
      async_ld_b64(off, g);
    }
  };

  issue_tile(0, 0);
  issue_tile(1, 1);

  for (int tl = 0; tl < NTILES; ++tl) {
    int buf = tl & 1;
    if (tl + 1 < NTILES) asm volatile("s_wait_asynccnt 4" ::: "memory");
    else                 asm volatile("s_wait_asynccnt 0" ::: "memory");

    #pragma unroll
    for (int j = 0; j < TILE; ++j) {
      int s = tl * TILE + j;
      double xv = tileS[buf][j * NTH + t];
      if (mode == 1) {
        int kk = (NEXT - 1 - s) - t * CHUNK;
        if (kk < 0) xv = 0.0;                             // beyond reversed stream
      }
      // direct-form II transposed step (matches reference's scan body)
      double y = bn[0] * xv + z[0];
      #pragma unroll
      for (int i = 0; i < ORD - 1; ++i)
        z[i] = bn[i + 1] * xv + z[i + 1] - an[i + 1] * y;
      z[ORD - 1] = bn[ORD] * xv - an[ORD] * y;

      if (WRITE) {
        if (mode == 0) {
          outRowT[(size_t)s * NTH + t] = y;               // coalesced f64 store
        } else {
          int kk = (NEXT - 1 - s) - t * CHUNK;            // y2 flat index -> out index
          int oi = kk - PAD;
          if (oi >= 0 && oi < NLEN)
            outF[(size_t)r * NLEN + oi] = (float)y;
        }
      }
    }
    if (tl + 2 < NTILES) issue_tile(tl + 2, buf);
  }
}

// ---------------------------------------------------------------------------
// Kernel 2/3: one row per block. Chunk-parallel exact linear-recurrence scan:
//   pass1 (zero-state residuals) -> sequential 8x8 fix-up -> pass2 (outputs)
// ---------------------------------------------------------------------------
__global__ __launch_bounds__(NTH) void iir_pass(const double* __restrict__ inT,
                                                const double* __restrict__ bcoef,
                                                const double* __restrict__ acoef,
                                                double* __restrict__ outT,
                                                float* __restrict__ outF,
                                                const float* __restrict__ x,
                                                int mode) {
  __shared__ double S[NTH][ORD];            // chunk states (v_c, then start states)
  __shared__ double Mm[ORD * ORD];          // M = A^CHUNK
  __shared__ double tileS[2][TILE * NTH];   // async staging buffers

  const int r = blockIdx.x;
  const int t = threadIdx.x;
  const double* inRow = inT + (size_t)r * TOT;
  double* outRowT = outT ? outT + (size_t)r * TOT : nullptr;

  // normalized taps
  double an[NC], bn[NC];
  {
    double a0 = acoef[0];
    #pragma unroll
    for (int i = 0; i < NC; ++i) { an[i] = acoef[i] / a0; bn[i] = bcoef[i] / a0; }
  }

  // --- M = A^CHUNK, one column per thread 0..7 (A = companion of the filter) ---
  if (t < ORD) {
    double c[ORD];
    #pragma unroll
    for (int i = 0; i < ORD; ++i) c[i] = (i == t) ? 1.0 : 0.0;
    for (int step = 0; step < CHUNK; ++step) {
      double c0 = c[0];
      #pragma unroll
      for (int i = 0; i < ORD - 1; ++i) c[i] = c[i + 1] - an[i + 1] * c0;
      c[ORD - 1] = -an[ORD] * c0;
    }
    #pragma unroll
    for (int i = 0; i < ORD; ++i) Mm[i * ORD + t] = c[i];
  }

  // --- zi (lfilter_zi) and x0 on thread 0: solve (I - A) zi = b[1:] - a[1:]*b0 ---
  double zi[ORD]; double x0 = 0.0;
  if (t == 0) {
    double G[ORD][ORD + 1];
    for (int i = 0; i < ORD; ++i) {
      for (int j = 0; j < ORD; ++j) {
        double g = (i == j) ? 1.0 : 0.0;
        if (j == 0)      g += an[i + 1];
        if (j == i + 1)  g -= 1.0;
        G[i][j] = g;
      }
      G[i][ORD] = bn[i + 1] - an[i + 1] * bn[0];
    }
    for (int c2 = 0; c2 < ORD; ++c2) {              // partial-pivot elimination
      int piv = c2; double best = fabs(G[c2][c2]);
      for (int rr = c2 + 1; rr < ORD; ++rr) {
        double v = fabs(G[rr][c2]);
        if (v > best) { best = v; piv = rr; }
      }
      if (piv != c2)
        for (int j = c2; j <= ORD; ++j) { double tmp = G[c2][j]; G[c2][j] = G[piv][j]; G[piv][j] = tmp; }
      double inv = 1.0 / G[c2][c2];
      for (int rr = c2 + 1; rr < ORD; ++rr) {
        double f = G[rr][c2] * inv;
        for (int j = c2; j <= ORD; ++j) G[rr][j] -= f * G[c2][j];
      }
    }
    for (int i = ORD - 1; i >= 0; --i) {
      double acc = G[i][ORD];
      for (int j = i + 1; j < ORD; ++j) acc -= G[i][j] * zi[j];
      zi[i] = acc / G[i][i];
    }
    if (mode == 0) {
      const float* xr = x + (size_t)r * NLEN;
      x0 = 2.0 * (double)xr[0] - (double)xr[PAD];   // ext[0]
    } else {
      int q0 = NEXT - 1;                            // first sample of reversed stream
      x0 = inRow[(size_t)(q0 % CHUNK) * NTH + (q0 / CHUNK)];
    }
  }

  // --- pass 1: zero-state residual per chunk ---
  double z[ORD];
  #pragma unroll
  for (int i = 0; i < ORD; ++i) z[i] = 0.0;
  run_chunk<false>(inRow, outRowT, outF, r, t, mode, an, bn, z, tileS);
  #pragma unroll
  for (int i = 0; i < ORD; ++i) S[t][i] = z[i];
  __syncthreads();

  // --- sequential fix-up: start(c+1) = M*start(c) + v_c (exact, linearity) ---
  if (t == 0) {
    double cur[ORD];
    #pragma unroll
    for (int i = 0; i < ORD; ++i) cur[i] = zi[i] * x0;
    for (int c = 0; c < NTH; ++c) {
      double v[ORD], nw[ORD];
      #pragma unroll
      for (int i = 0; i < ORD; ++i) v[i] = S[c][i];
      #pragma unroll
      for (int i = 0; i < ORD; ++i) S[c][i] = cur[i];   // start state for chunk c
      #pragma unroll
      for (int i = 0; i < ORD; ++i) {
        double acc = v[i];
        #pragma unroll
        for (int j = 0; j < ORD; ++j) acc += Mm[i * ORD + j] * cur[j];
        nw[i] = acc;
      }
      #pragma unroll
      for (int i = 0; i < ORD; ++i) cur[i] = nw[i];
    }
  }
  __syncthreads();

  // --- pass 2: re-run with exact start state, emit outputs ---
  #pragma unroll
  for (int i = 0; i < ORD; ++i) z[i] = S[t][i];
  run_chunk<true>(inRow, outRowT, outF, r, t, mode, an, bn, z, tileS);
}

// ---------------------------------------------------------------------------
extern "C" void kernel_launch(void* const* d_in, const int* in_sizes, int n_in,
                              void* d_out, int out_size, void* d_ws, size_t ws_size,
                              hipStream_t stream) {
  (void)in_sizes; (void)n_in; (void)out_size; (void)ws_size;
  const float*  x = (const float*)d_in[0];
  const double* b = (const double*)d_in[1];
  const double* a = (const double*)d_in[2];
  float* out = (float*)d_out;

  double* extT = (double*)d_ws;                       // [NROWS][TOT] f64
  double* y1T  = extT + (size_t)NROWS * TOT;          // [NROWS][TOT] f64

  size_t tot = (size_t)NROWS * TOT;
  build_extT<<<(unsigned)((tot + 255) / 256), 256, 0, stream>>>(x, extT);
  // pass A: forward filter of the extended signal
  iir_pass<<<NROWS, NTH, 0, stream>>>(extT, b, a, y1T, nullptr, x, 0);
  // pass B: filter the reversed stream, scatter final f32 output
  iir_pass<<<NROWS, NTH, 0, stream>>>(y1T,  b, a, nullptr, out, x, 1);
}